// VectorQuantizer_4037269259120
// MI455X (gfx1250) — compile-verified
//
#include <hip/hip_runtime.h>
#include <hip/hip_bf16.h>

typedef __attribute__((ext_vector_type(2))) float v2f;
typedef __attribute__((ext_vector_type(8))) float v8f;

#define BATCH    64
#define CHAN     256
#define HW       1024          // 32*32
#define NCODES   512
#define NROWS    (BATCH * HW)  // 65536
#define ROWS_BLK 16
#define NBLOCKS  (NROWS / ROWS_BLK)  // 4096
#define TILES_PER_WAVE 4       // 512 codes / 16 / 8 waves

// ---------------------------------------------------------------------------
// Kernel 1: codebook squared norms ||c_j||^2  (512 codes x 256 dims)
// ---------------------------------------------------------------------------
__global__ void vq_cnorm_kernel(const float* __restrict__ cb,
                                float* __restrict__ cnorm) {
  int i = blockIdx.x * blockDim.x + threadIdx.x;
  if (i < NCODES) {
    const float* row = cb + (size_t)i * CHAN;
    float s = 0.f;
#pragma unroll 8
    for (int k = 0; k < CHAN; ++k) { float v = row[k]; s += v * v; }
    cnorm[i] = s;
  }
}

// ---------------------------------------------------------------------------
// Kernel 2: main VQ kernel. One 256-thread block per 16 rows.
//   - stage z tile transposed into LDS ([k][m], 16KB)
//   - WMMA f32 16x16x4: scores S = z_tile @ cb^T, 8 waves x 4 N-tiles
//   - argmax(2S - ||c||^2) via shfl_xor butterflies + cross-wave LDS merge
//   - gather z_q, write [B,C,H,W] output, per-block loss partial to ws
// ---------------------------------------------------------------------------
__launch_bounds__(256, 2)
__global__ void vq_main_kernel(const float* __restrict__ z,
                               const float* __restrict__ cb,
                               const float* __restrict__ cnorm,
                               float* __restrict__ out_zq,
                               float* __restrict__ out_idx,
                               float* __restrict__ loss_partials) {
  __shared__ float zTile[CHAN * ROWS_BLK];   // [k*16 + m], 16 KB
  __shared__ float sBest[8 * 16];            // [wave][row]
  __shared__ int   sBestIdx[8 * 16];
  __shared__ int   sFinalIdx[16];
  __shared__ float sRed[8];

  const int tid  = threadIdx.x;
  const int wave = tid >> 5;
  const int lane = tid & 31;
  const int half = lane >> 4;   // K-pair selector inside the wave
  const int l16  = lane & 15;

  const int rowBase = blockIdx.x * ROWS_BLK;   // flat row = b*HW + hw
  const int b   = rowBase / HW;
  const int hw0 = rowBase % HW;                // multiple of 16

  // ---- stage z tile: zTile[k*16+m] = z[b, k, hw0+m], coalesced global reads
  {
    const int m  = tid & 15;
    const int kb = tid >> 4;                   // 0..15
    const float* zp = z + (size_t)b * CHAN * HW + (size_t)hw0 + m;
#pragma unroll
    for (int j = 0; j < 16; ++j) {
      const int k = kb + j * 16;
      zTile[k * 16 + m] = zp[(size_t)k * HW];
    }
  }
  __syncthreads();

  // ---- WMMA accumulation: wave owns codes [wave*64, wave*64+64)
  v8f acc[TILES_PER_WAVE] = {};
  const int nBase = wave * (TILES_PER_WAVE * 16);

  for (int k0 = 0; k0 < CHAN; k0 += 4) {
    const int ka = k0 + 2 * half;
    v2f a;
    a.x = zTile[(ka    ) * 16 + l16];
    a.y = zTile[(ka + 1) * 16 + l16];
#pragma unroll
    for (int t = 0; t < TILES_PER_WAVE; ++t) {
      const float* cp = cb + (size_t)(nBase + t * 16 + l16) * CHAN + ka;
      v2f bm; bm.x = cp[0]; bm.y = cp[1];
      // D = A(16x4) x B(4x16) + C : v_wmma_f32_16x16x4_f32
      acc[t] = __builtin_amdgcn_wmma_f32_16x16x4_f32(
          false, a, false, bm, (short)0, acc[t], false, false);
    }
  }

  // ---- per-row argmax of (2S - ||c||^2) over this wave's 64 codes
  // vgpr r of acc: row m = r + 8*half, code n = nBase + t*16 + l16
#pragma unroll
  for (int r = 0; r < 8; ++r) {
    float bs = -3.402823e38f; int bi = 0x7fffffff;
#pragma unroll
    for (int t = 0; t < TILES_PER_WAVE; ++t) {
      const int n = nBase + t * 16 + l16;
      const float s = 2.f * acc[t][r] - cnorm[n];
      if (s > bs || (s == bs && n < bi)) { bs = s; bi = n; }
    }
#pragma unroll
    for (int mk = 1; mk < 16; mk <<= 1) {   // stays within 16-lane half
      const float os = __shfl_xor(bs, mk, 32);
      const int   oi = __shfl_xor(bi, mk, 32);
      if (os > bs || (os == bs && oi < bi)) { bs = os; bi = oi; }
    }
    if (l16 == 0) {
      const int m = r + 8 * half;
      sBest[wave * 16 + m]    = bs;
      sBestIdx[wave * 16 + m] = bi;
    }
  }
  __syncthreads();

  // ---- merge across 8 waves, tie-break lowest index (matches jnp.argmin)
  if (tid < 16) {
    const int m = tid;
    float bs = sBest[m]; int bi = sBestIdx[m];
#pragma unroll
    for (int w = 1; w < 8; ++w) {
      const float s = sBest[w * 16 + m];
      const int   i2 = sBestIdx[w * 16 + m];
      if (s > bs || (s == bs && i2 < bi)) { bs = s; bi = i2; }
    }
    sFinalIdx[m] = bi;
    out_idx[rowBase + m] = (float)bi;
  }
  __syncthreads();

  // ---- phase 2: gather codebook rows, write z_q_st in [B,C,H,W], loss
  float lsum = 0.f;
  {
    const int m  = tid & 15;
    const int kb = tid >> 4;
    const int idx = sFinalIdx[m];
    const float* crow = cb + (size_t)idx * CHAN;
    const size_t obase = (size_t)b * CHAN * HW + (size_t)hw0 + m;
#pragma unroll
    for (int j = 0; j < 16; ++j) {
      const int k = kb + j * 16;
      const float q  = crow[k];
      const float zv = zTile[k * 16 + m];
      const float d  = q - zv;
      lsum += d * d;
      out_zq[obase + (size_t)k * HW] = q;
    }
  }
#pragma unroll
  for (int mk = 16; mk >= 1; mk >>= 1) lsum += __shfl_xor(lsum, mk, 32);
  if (lane == 0) sRed[wave] = lsum;
  __syncthreads();
  if (tid == 0) {
    float tot = 0.f;
#pragma unroll
    for (int w = 0; w < 8; ++w) tot += sRed[w];
    loss_partials[blockIdx.x] = tot;   // deterministic: no float atomics
  }
}

// ---------------------------------------------------------------------------
// Kernel 3: deterministic loss reduction over 4096 block partials
// loss = (0.25 + 1.0) * mean((z_q - z)^2)
// ---------------------------------------------------------------------------
__global__ void vq_loss_reduce_kernel(const float* __restrict__ partials,
                                      float* __restrict__ out_loss) {
  __shared__ float sRed[8];
  float s = 0.f;
  for (int i = threadIdx.x; i < NBLOCKS; i += 256) s += partials[i];
#pragma unroll
  for (int mk = 16; mk >= 1; mk >>= 1) s += __shfl_xor(s, mk, 32);
  const int wave = threadIdx.x >> 5, lane = threadIdx.x & 31;
  if (lane == 0) sRed[wave] = s;
  __syncthreads();
  if (threadIdx.x == 0) {
    float t = 0.f;
#pragma unroll
    for (int w = 0; w < 8; ++w) t += sRed[w];
    *out_loss = t * (1.25f / (float)((size_t)NROWS * CHAN));
  }
}

// ---------------------------------------------------------------------------
extern "C" void kernel_launch(void* const* d_in, const int* in_sizes, int n_in,
                              void* d_out, int out_size, void* d_ws, size_t ws_size,
                              hipStream_t stream) {
  const float* z  = (const float*)d_in[0];   // [64,256,32,32]
  const float* cb = (const float*)d_in[1];   // [512,256]

  float* out      = (float*)d_out;
  float* out_zq   = out;                                   // 16777216 f32
  float* out_idx  = out + (size_t)NROWS * CHAN;            // 65536 (as float)
  float* out_loss = out + (size_t)NROWS * CHAN + NROWS;    // 1

  float* cnorm    = (float*)d_ws;            // 512 f32
  float* partials = cnorm + NCODES;          // 4096 f32

  vq_cnorm_kernel<<<2, 256, 0, stream>>>(cb, cnorm);
  vq_main_kernel<<<NBLOCKS, 256, 0, stream>>>(z, cb, cnorm,
                                              out_zq, out_idx, partials);
  vq_loss_reduce_kernel<<<1, 256, 0, stream>>>(partials, out_loss);
}